// Centroid_70325794505046
// MI455X (gfx1250) — compile-verified
//
#include <hip/hip_runtime.h>
#include <hip/hip_bf16.h>

// Centroid EMA update: out = THETA*centroid + (1-THETA)*segsum(embed,y)/(count+eps)
// Strategy: counting-sort rows by class, then block-sparse one-hot matmul with
// exact-fp32 V_WMMA_F32_16X16X4_F32 over sorted row buckets. Memory-bound
// (~72 MiB -> ~3.1 us @ 23.3 TB/s); zero float atomics in the hot pass.
// Per-class scale 0.3/(count+eps) is precomputed once (1000 divides total),
// so the hot epilogue is load + FMA only.

#define NUM_CLASSES 1000
#define EMBED_DIM   1024
#define BATCH       16384
#define THETA       0.7f
#define OMTHETA     0.3f
#define EPS         1e-8f

#define CGROUPS 63          // ceil(1000/16) groups of 16 classes
#define WPB     8           // waves per block (256 threads)

typedef __attribute__((ext_vector_type(2))) float v2f;
typedef __attribute__((ext_vector_type(8))) float v8f;

// ---------------- sort passes (negligible cost) ----------------

__global__ void k_zero(int* __restrict__ counts) {
    counts[threadIdx.x] = 0;                       // 1 block x 1024
}

__global__ void k_hist(const int* __restrict__ y, int* __restrict__ counts) {
    int i = blockIdx.x * blockDim.x + threadIdx.x;
    if (i < BATCH) atomicAdd(&counts[y[i]], 1);
}

__global__ void k_scan(const int* __restrict__ counts,
                       int* __restrict__ start, int* __restrict__ cursor,
                       float* __restrict__ fscale) {
    __shared__ int buf[1024];
    int t = threadIdx.x;
    int c0 = counts[t];
    buf[t] = c0;
    __syncthreads();
    for (int off = 1; off < 1024; off <<= 1) {
        int v = (t >= off) ? buf[t - off] : 0;
        __syncthreads();
        buf[t] += v;
        __syncthreads();
    }
    int excl = buf[t] - c0;                        // exclusive prefix sum
    start[t]  = excl;                              // classes >=1000 have count 0
    cursor[t] = excl;                              // -> start saturates at BATCH
    fscale[t] = OMTHETA / ((float)c0 + EPS);       // per-class scale, done once
}

__global__ void k_scatter(const int* __restrict__ y,
                          int* __restrict__ cursor, int* __restrict__ sorted_row) {
    int i = blockIdx.x * blockDim.x + threadIdx.x;
    if (i < BATCH) {
        int pos = atomicAdd(&cursor[y[i]], 1);
        sorted_row[pos] = i;
    }
}

// ---------------- main pass: block-sparse one-hot WMMA ----------------
// grid = CGROUPS*WPB blocks of 256; one wave owns one (class-group, 16-dim tile).

__global__ __launch_bounds__(256)
void k_centroid_wmma(const float* __restrict__ embed,
                     const float* __restrict__ centroid,
                     const int*   __restrict__ start,
                     const int*   __restrict__ sorted_row,
                     const float* __restrict__ fscale,
                     float*       __restrict__ out) {
    const int wave   = threadIdx.x >> 5;
    const int lane   = threadIdx.x & 31;
    const int group  = blockIdx.x >> 3;                  // 0..62
    const int ntile  = ((blockIdx.x & 7) << 3) + wave;   // 0..63
    const int cbase  = group << 4;
    const int dimb   = ntile << 4;

    const int mA   = lane & 15;            // A-matrix row for this lane
    const int kk   = (lane >> 4) << 1;     // K-slots held by this lane: kk, kk+1
    const int n    = dimb + mA;            // B/D column owned by this lane

    // class position range for this lane's A-row (rows sorted by class)
    const int s0   = start[cbase + mA];
    const int s1   = start[cbase + mA + 1];
    const int gbeg = start[cbase];         // uniform across wave
    const int gend = start[cbase + 16];    // uniform across wave
    const int last = gend - 1;

    // 4 independent accumulator chains -> 4 in-flight gather+WMMA streams
    v8f accs[4];
    #pragma unroll
    for (int u = 0; u < 4; ++u) accs[u] = (v8f){};

    for (int p = gbeg; p < gend; p += 16) {
        #pragma unroll
        for (int u = 0; u < 4; ++u) {
            int q  = p + 4 * u;
            int i0 = q + kk, i1 = i0 + 1;

            // exact 0/1 indicator: position belongs to class cbase+mA ?
            v2f a, b;
            a.x = (i0 >= s0 && i0 < s1) ? 1.0f : 0.0f;
            a.y = (i1 >= s0 && i1 < s1) ? 1.0f : 0.0f;

            // gather 2 row slices (clamped; clamped lanes are masked by A==0)
            int r0 = sorted_row[min(i0, last)];
            int r1 = sorted_row[min(i1, last)];
            b.x = embed[r0 * EMBED_DIM + n];
            b.y = embed[r1 * EMBED_DIM + n];

            // D(16x16) += A(16x4) x B(4x16), exact fp32
            accs[u] = __builtin_amdgcn_wmma_f32_16x16x4_f32(
                          false, a, false, b, (short)0, accs[u], false, false);
        }
    }
    v8f acc = (accs[0] + accs[1]) + (accs[2] + accs[3]);

    // epilogue: D layout -> VGPR v holds M = v + 8*(lane>=16), N = lane&15
    const int hi = lane >> 4;
    if (cbase + 16 <= NUM_CLASSES) {
        // uniform fast path: all 16 classes valid (62 of 63 groups)
        #pragma unroll
        for (int v = 0; v < 8; ++v) {
            int c   = cbase + v + (hi << 3);
            int idx = c * EMBED_DIM + n;
            out[idx] = THETA * centroid[idx] + acc[v] * fscale[c];
        }
    } else {
        #pragma unroll
        for (int v = 0; v < 8; ++v) {
            int c = cbase + v + (hi << 3);
            if (c < NUM_CLASSES) {
                int idx = c * EMBED_DIM + n;
                out[idx] = THETA * centroid[idx] + acc[v] * fscale[c];
            }
        }
    }
}

// ---------------- launch ----------------

extern "C" void kernel_launch(void* const* d_in, const int* in_sizes, int n_in,
                              void* d_out, int out_size, void* d_ws, size_t ws_size,
                              hipStream_t stream) {
    const float* embed    = (const float*)d_in[0];
    const int*   y        = (const int*)  d_in[1];
    const float* centroid = (const float*)d_in[2];
    float*       out      = (float*)d_out;

    int*   ws     = (int*)d_ws;
    int*   counts = ws;                 // 1024 ints
    int*   start  = ws + 1024;          // 1024 ints (exclusive prefix, saturates at BATCH)
    int*   cursor = ws + 2048;          // 1024 ints
    float* fscale = (float*)(ws + 3072);// 1024 floats: 0.3/(count+eps)
    int*   sorted = ws + 4096;          // BATCH ints

    k_zero   <<<1,          1024, 0, stream>>>(counts);
    k_hist   <<<BATCH/256,  256,  0, stream>>>(y, counts);
    k_scan   <<<1,          1024, 0, stream>>>(counts, start, cursor, fscale);
    k_scatter<<<BATCH/256,  256,  0, stream>>>(y, cursor, sorted);
    k_centroid_wmma<<<CGROUPS * WPB, 256, 0, stream>>>(embed, centroid, start, sorted,
                                                       fscale, out);
}